// FinElteScheme_30640296690422
// MI455X (gfx1250) — compile-verified
//
#include <hip/hip_runtime.h>

// Problem constants (from reference): S=256, T=128, n = NX-2 = 1024.
#define NS 256
#define NT 128
#define NN 1024
#define DIFF 0.17f   // (VOL^2 + VOL0^2)/2 = (0.25 + 0.09)/2

typedef __attribute__((ext_vector_type(2))) float v2f;
typedef __attribute__((ext_vector_type(8))) float v8f;

// Wave32 sum of (a0 + a1) over all 32 lanes, via V_WMMA_F32_16X16X4_F32.
// A(16x4) holds the 64 lane partials (lane L<16: row L gets k=0,1; lane L>=16:
// row L-16 gets k=2,3). B = ones => D[m][n] = rowsum(A,m) for every n.
// D VGPR v: lanes 0-15 = S_v, lanes 16-31 = S_{v+8}; summing the 8 D regs and
// adding the opposite half (shfl_xor 16) yields the full wave sum in all lanes.
__device__ __forceinline__ float wmma_wave_sum(float a0, float a1) {
  v2f A; A[0] = a0;   A[1] = a1;
  v2f B; B[0] = 1.0f; B[1] = 1.0f;
  v8f C = {};
  C = __builtin_amdgcn_wmma_f32_16x16x4_f32(false, A, false, B, (short)0, C,
                                            false, false);
  float r = ((C[0] + C[1]) + (C[2] + C[3])) + ((C[4] + C[5]) + (C[6] + C[7]));
  r += __shfl_xor(r, 16, 32);
  return r;
}

// One block per sample; full time loop in-kernel. PCR tridiagonal solve in LDS.
__global__ __launch_bounds__(256) void fem_mckean_kernel(
    const float* __restrict__ mt, const float* __restrict__ msp,
    const float* __restrict__ ic, const float* __restrict__ noise,
    float* __restrict__ density, float* __restrict__ wsRun,
    float* __restrict__ wsLast) {
  const int s    = blockIdx.x;
  const int tid  = threadIdx.x;
  const int lane = tid & 31;
  const int wv   = tid >> 5;

  __shared__ float rhoL[NN + 2];        // rho with zero halo
  __shared__ float bufA[2][NN];         // PCR ping-pong: sub-diag
  __shared__ float bufB[2][NN];         // diag
  __shared__ float bufC[2][NN];         // super-diag
  __shared__ float bufD[2][NN];         // rhs
  __shared__ float red[8][4];           // per-wave reduction scratch

  const float dx  = msp[1] - msp[0];
  const float dt  = mt[1] - mt[0];      // linspace -> constant dt
  const float dx6 = dx * (1.0f / 6.0f);
  const float Ddx = DIFF / dx;
  const float h   = 0.5f * dt;

  // ---- per-thread, time-invariant point data (4 stride-256 points) ----
  int   jj[4];
  float xj[4], wI[4], wC[4], aB[4], cB[4], bD[4];
#pragma unroll
  for (int q = 0; q < 4; ++q) {
    const int j = tid + 256 * q;
    jj[q] = j;
    const float x   = -3.0f + (float)(j + 1) * dx;  // x_in[j] = mesh[j+1]
    const float xm  = -3.0f + (float)(j)     * dx;  // mesh[j]   (x_{j-1})
    const float xn  = -3.0f + (float)(j + 2) * dx;  // mesh[j+2] (x_{j+1})
    const float Ix  = fmaxf(-x,  0.0f);             // intensity, IP = 1
    const float Ixm = fmaxf(-xm, 0.0f);
    const float Ixn = fmaxf(-xn, 0.0f);
    // abs0_j = dx/6 * e_{max(j-1,0)},  e_k = (I(x_k)+I(x_{k+1}))/2
    const float abs0 = (j == 0) ? dx6 * 0.5f * (Ix + Ixn)
                                : dx6 * 0.5f * (Ixm + Ix);
    const float abs1 = dx6 * (Ixm + 2.0f * Ix + Ixn);
    // d1 is sample-independent: (coeff_j - coeff_{j+1})/2; K cancels.
    const float d1 = (j < NN - 1) ? (-0.5f * dx) : (0.5f * (0.5f * x - 1.5f));
    xj[q] = x;
    wI[q] = Ix;
    wC[q] = x * x;
    bD[q] = 4.0f * dx6 + dt * (abs1 + d1 + 2.0f * Ddx);           // diagonal
    aB[q] = (j == 0) ? 0.0f : (dx6 - dt * Ddx + h * x);           // + h*K
    cB[q] = (j == NN - 1) ? (dt * abs0)
                          : (dx6 + dt * abs0 - dt * Ddx - h * xn); // - h*K
  }

  // ---- init: rho0 = ic, density[s,0,:] = ic ----
#pragma unroll
  for (int q = 0; q < 4; ++q) {
    const int j   = jj[q];
    const float v = ic[j];
    rhoL[1 + j] = v;
    density[(size_t)s * NT * NN + j] = v;
  }
  if (tid == 0) { rhoL[0] = 0.0f; rhoL[NN + 1] = 0.0f; }
  float costRun = 0.0f;
  __syncthreads();

  for (int t = 0; t < NT - 1; ++t) {
    // --- integrals of rho_t: intensity, mean, running-cost (WMMA reduce) ---
    float pI0 = 0.f, pI1 = 0.f, pM0 = 0.f, pM1 = 0.f, pC0 = 0.f, pC1 = 0.f;
#pragma unroll
    for (int q = 0; q < 2; ++q) {
      const float r = rhoL[1 + jj[q]];
      pI0 += wI[q] * r; pM0 += xj[q] * r; pC0 += wC[q] * r;
    }
#pragma unroll
    for (int q = 2; q < 4; ++q) {
      const float r = rhoL[1 + jj[q]];
      pI1 += wI[q] * r; pM1 += xj[q] * r; pC1 += wC[q] * r;
    }
    const float wSI = wmma_wave_sum(pI0, pI1);
    const float wSM = wmma_wave_sum(pM0, pM1);
    const float wSC = wmma_wave_sum(pC0, pC1);
    if (lane == 0) { red[wv][0] = wSI; red[wv][1] = wSM; red[wv][2] = wSC; }
    __syncthreads();
    float sI = 0.f, sM = 0.f, sC = 0.f;
#pragma unroll
    for (int w = 0; w < 8; ++w) { sI += red[w][0]; sM += red[w][1]; sC += red[w][2]; }
    costRun += sC;  // sum_j x^2 * rho_t (scaled at the end)

    const float Ival = dx * sI;
    const float mean = dx * sM;
    const float fb   = Ival + 0.5f * Ival * Ival;        // FB_P = 1
    const float K    = fb - 0.5f * mean;                 // coeff_j = x_j + K
    const float sc   = 0.15f * noise[s * (NT - 1) + t];  // VOL0*dW/2

    // --- assemble tridiagonal system + rhs into PCR buffer 0 ---
#pragma unroll
    for (int q = 0; q < 4; ++q) {
      const int j    = jj[q];
      const float rm = rhoL[j];        // rho_{j-1} (halo zero)
      const float r0 = rhoL[j + 1];    // rho_j
      const float rp = rhoL[j + 2];    // rho_{j+1} (halo zero)
      bufA[0][j] = (j == 0)      ? 0.0f  : (aB[q] + h * K);
      bufB[0][j] = bD[q];
      bufC[0][j] = (j == NN - 1) ? cB[q] : (cB[q] - h * K);
      bufD[0][j] = dx6 * (rm + 4.0f * r0 + rp) + sc * (rm - rp);
    }
    __syncthreads();

    // --- Parallel Cyclic Reduction: 10 rounds, strides 1..512 ---
    int p = 0;
    for (int k = 1; k < NN; k <<= 1) {
#pragma unroll
      for (int q = 0; q < 4; ++q) {
        const int i    = jj[q];
        const float ai = bufA[p][i], bi = bufB[p][i];
        const float ci = bufC[p][i], di = bufD[p][i];
        float na = 0.0f, nc = 0.0f, nb = bi, nd = di;
        if (i - k >= 0) {
          const float alpha = ai / bufB[p][i - k];
          na  = -alpha * bufA[p][i - k];
          nb -=  alpha * bufC[p][i - k];
          nd -=  alpha * bufD[p][i - k];
        }
        if (i + k < NN) {
          const float gamma = ci / bufB[p][i + k];
          nc  = -gamma * bufC[p][i + k];
          nb -=  gamma * bufA[p][i + k];
          nd -=  gamma * bufD[p][i + k];
        }
        bufA[p ^ 1][i] = na; bufB[p ^ 1][i] = nb;
        bufC[p ^ 1][i] = nc; bufD[p ^ 1][i] = nd;
      }
      __syncthreads();
      p ^= 1;
    }

    // --- decoupled solve, advance rho, emit trajectory row ---
#pragma unroll
    for (int q = 0; q < 4; ++q) {
      const int i      = jj[q];
      const float xnew = bufD[p][i] / bufB[p][i];
      rhoL[1 + i] = xnew;
      density[(size_t)s * NT * NN + (size_t)(t + 1) * NN + i] = xnew;
    }
    __syncthreads();
  }

  // --- terminal cost partial: sum_j x^2 * rho_T ---
  float pL0 = 0.f, pL1 = 0.f;
#pragma unroll
  for (int q = 0; q < 2; ++q) pL0 += wC[q] * rhoL[1 + jj[q]];
#pragma unroll
  for (int q = 2; q < 4; ++q) pL1 += wC[q] * rhoL[1 + jj[q]];
  const float wSL = wmma_wave_sum(pL0, pL1);
  if (lane == 0) red[wv][3] = wSL;
  __syncthreads();
  if (tid == 0) {
    float sL = 0.f;
#pragma unroll
    for (int w = 0; w < 8; ++w) sL += red[w][3];
    wsRun[s]  = costRun;
    wsLast[s] = sL;
  }
}

// Deterministic cross-sample cost reduction (single block).
__global__ __launch_bounds__(256) void fem_finalize_kernel(
    const float* __restrict__ mt, const float* __restrict__ msp,
    const float* __restrict__ wsRun, const float* __restrict__ wsLast,
    float* __restrict__ cost) {
  __shared__ float r0[256];
  __shared__ float r1[256];
  const int tid = threadIdx.x;
  r0[tid] = wsRun[tid];
  r1[tid] = wsLast[tid];
  __syncthreads();
  for (int off = 128; off > 0; off >>= 1) {
    if (tid < off) { r0[tid] += r0[tid + off]; r1[tid] += r1[tid + off]; }
    __syncthreads();
  }
  if (tid == 0) {
    const float dx = msp[1] - msp[0];
    const float dt = mt[1] - mt[0];
    // costs.sum() = dt*dx*0.125/S * sum_{s,t,j} x^2 rho ; running_cost(-x/2)=x^2/8
    cost[0] = dt * dx * 0.125f * (1.0f / (float)NS) * r0[0];
    cost[1] = dx * (1.0f / (float)NS) * r1[0];
  }
}

__global__ void fem_copy_noise_kernel(const float* __restrict__ noise,
                                      float* __restrict__ out, int n) {
  const int i = blockIdx.x * blockDim.x + threadIdx.x;
  if (i < n) out[i] = noise[i];
}

extern "C" void kernel_launch(void* const* d_in, const int* in_sizes, int n_in,
                              void* d_out, int out_size, void* d_ws,
                              size_t ws_size, hipStream_t stream) {
  (void)in_sizes; (void)n_in; (void)out_size; (void)ws_size;
  const float* mt    = (const float*)d_in[0];  // mesh_time (128)
  const float* msp   = (const float*)d_in[1];  // mesh_space (1026)
  const float* ic    = (const float*)d_in[2];  // initial_condition (1024)
  const float* noise = (const float*)d_in[3];  // common_noise (256*127)

  float* out     = (float*)d_out;
  float* density = out;                                   // (256,128,1024)
  float* cost    = out + (size_t)NS * NT * NN;            // (2,)
  float* nz_out  = cost + 2;                              // (256,127)
  float* ws      = (float*)d_ws;                          // 2*256 floats

  fem_mckean_kernel<<<NS, 256, 0, stream>>>(mt, msp, ic, noise, density,
                                            ws, ws + NS);
  fem_finalize_kernel<<<1, 256, 0, stream>>>(mt, msp, ws, ws + NS, cost);
  const int nn = NS * (NT - 1);
  fem_copy_noise_kernel<<<(nn + 255) / 256, 256, 0, stream>>>(noise, nz_out, nn);
}